// GaussianFPSPooling_14568529068105
// MI455X (gfx1250) — compile-verified
//
#include <hip/hip_runtime.h>

#define FPS_K   256
#define IN_D    128
#define OUT_D   256
#define FPS_T   1024   // 32 wave32 waves per workgroup
#define WAVES   (FPS_T / 32)

typedef __attribute__((ext_vector_type(2))) float v2f;
typedef __attribute__((ext_vector_type(8))) float v8f;

struct Pt3 { float x, y, z; };   // 12-byte -> global_load_b96

// ---------------------------------------------------------------------------
// Phase 1: farthest point sampling. One persistent workgroup per batch.
// dists[b][i] in global workspace (400KB/batch > 320KB LDS; all L2-resident).
// Argmax per step: wave32 shuffle butterfly + single 32-entry LDS pass.
// Only 2 workgroup barriers per step (vs 11 for a full LDS tree).
// ---------------------------------------------------------------------------
__global__ __launch_bounds__(FPS_T) void fps_kernel(
    const float* __restrict__ means, int N,
    float* __restrict__ dists, int* __restrict__ idx_out)
{
    const int b    = blockIdx.x;
    const int t    = threadIdx.x;
    const int wave = t >> 5;
    const int lane = t & 31;

    const Pt3*  m   = (const Pt3*)(means + (size_t)b * N * 3);
    float*      dst = dists + (size_t)b * N;

    __shared__ float s_d[WAVES];
    __shared__ int   s_i[WAVES];
    __shared__ int   s_last;

    for (int i = t; i < N; i += FPS_T) dst[i] = 3.402823466e38f;
    if (t == 0) s_last = 0;
    __syncthreads();

    for (int k = 0; k < FPS_K; ++k) {
        const int last = s_last;
        if (t == 0) idx_out[(size_t)b * FPS_K + k] = last;

        const Pt3 p = m[last];

        float best  = -1.0f;
        int   besti = 0;
        for (int i = t; i < N; i += FPS_T) {
            const Pt3 q = m[i];
            const float dx = q.x - p.x;
            const float dy = q.y - p.y;
            const float dz = q.z - p.z;
            const float d2 = fmaf(dx, dx, fmaf(dy, dy, dz * dz));
            const float nd = fminf(dst[i], d2);
            dst[i] = nd;
            if (nd > best) { best = nd; besti = i; }   // lowest i among maxima
        }

        // wave32 butterfly reduce: (max dist, min index) — associative, no barriers
        #pragma unroll
        for (int off = 16; off > 0; off >>= 1) {
            const float od = __shfl_xor(best, off, 32);
            const int   oi = __shfl_xor(besti, off, 32);
            if (od > best || (od == best && oi < besti)) { best = od; besti = oi; }
        }
        if (lane == 0) { s_d[wave] = best; s_i[wave] = besti; }
        __syncthreads();

        // wave 0 reduces the 32 per-wave winners
        if (wave == 0) {
            float wd = s_d[lane];
            int   wi = s_i[lane];
            #pragma unroll
            for (int off = 16; off > 0; off >>= 1) {
                const float od = __shfl_xor(wd, off, 32);
                const int   oi = __shfl_xor(wi, off, 32);
                if (od > wd || (od == wd && oi < wi)) { wd = od; wi = oi; }
            }
            if (lane == 0) s_last = wi;
        }
        __syncthreads();
    }
}

// ---------------------------------------------------------------------------
// Phase 2: gather sampled rows + GEMM via V_WMMA_F32_16X16X4_F32.
// One wave per 16x16 output tile; 32 chained WMMAs over K=128.
//
// Fragment layouts (CDNA5 ISA 7.12.2, wave32):
//   A 16x4 f32 : lanes 0-15 -> M=lane, VGPR0=K0,VGPR1=K1; lanes 16-31 -> K2,K3
//   B 4x16 f32 : lanes 0-15 -> N=lane, VGPR0=K0,VGPR1=K1; lanes 16-31 -> K2,K3
//   C/D 16x16  : VGPR r, lanes 0-15 -> M=r, lanes 16-31 -> M=r+8; N=lane&15
// ---------------------------------------------------------------------------
__global__ __launch_bounds__(256) void gemm_wmma_kernel(
    const float* __restrict__ feats, const int* __restrict__ idx,
    const float* __restrict__ W, const float* __restrict__ bias,
    float* __restrict__ out, int N)
{
    const int wave = threadIdx.x >> 5;
    const int lane = threadIdx.x & 31;
    const int tile = blockIdx.x * 8 + wave;        // 1024 tiles total

    const int b  = tile >> 8;                      // batch
    const int tm = (tile >> 4) & 15;               // row-tile
    const int tn = tile & 15;                      // col-tile

    const int lhalf = lane >> 4;                   // 0 or 1
    const int l15   = lane & 15;

    const int row = idx[b * FPS_K + tm * 16 + l15];
    const float* arow = feats + ((size_t)b * N + row) * IN_D;
    const float* wcol = W + tn * 16 + l15;

    v8f acc = {};
#pragma unroll
    for (int k0 = 0; k0 < IN_D; k0 += 4) {
        const int ka = k0 + lhalf * 2;
        v2f a, bf;
        a.x  = arow[ka];
        a.y  = arow[ka + 1];
        bf.x = wcol[(size_t)ka * OUT_D];
        bf.y = wcol[(size_t)(ka + 1) * OUT_D];
        acc = __builtin_amdgcn_wmma_f32_16x16x4_f32(
            /*neg_a=*/false, a, /*neg_b=*/false, bf,
            /*c_mod=*/(short)0, acc, /*reuse_a=*/false, /*reuse_b=*/false);
    }

    const int n  = tn * 16 + l15;
    const int m0 = tm * 16 + lhalf * 8;
    const float bv = bias[n];
    float* orow = out + ((size_t)b * FPS_K + m0) * OUT_D + n;
#pragma unroll
    for (int r = 0; r < 8; ++r)
        orow[(size_t)r * OUT_D] = acc[r] + bv;
}

// ---------------------------------------------------------------------------
extern "C" void kernel_launch(void* const* d_in, const int* in_sizes, int n_in,
                              void* d_out, int out_size, void* d_ws, size_t ws_size,
                              hipStream_t stream) {
    const float* feats = (const float*)d_in[0];   // [B, N, 128] f32
    const float* means = (const float*)d_in[1];   // [B, N, 3]   f32
    const float* W     = (const float*)d_in[2];   // [128, 256]  f32
    const float* bias  = (const float*)d_in[3];   // [256]       f32
    float*       out   = (float*)d_out;           // [B, 256, 256] f32

    const int B = 4;
    const int N = in_sizes[1] / (B * 3);

    float* dists = (float*)d_ws;                                  // B*N floats
    int*   idx   = (int*)((char*)d_ws + (size_t)B * N * sizeof(float)); // B*256 ints

    fps_kernel<<<B, FPS_T, 0, stream>>>(means, N, dists, idx);

    const int tiles = B * (FPS_K / 16) * (OUT_D / 16);            // 1024
    gemm_wmma_kernel<<<tiles / 8, 256, 0, stream>>>(feats, idx, W, bias, out, N);
}